// R2d2Net_11914239279663
// MI455X (gfx1250) — compile-verified
//
#include <hip/hip_runtime.h>
#include <hip/hip_bf16.h>
#include <stdint.h>

// ---------------- problem constants ----------------
#define C_DIM   128
#define H_DIM   768
#define W_DIM   1024
#define HW      (H_DIM * W_DIM)        // 786432
#define K_OUT   5000
#define NBINS   4096
#define MAXCAND 262144
#define NSORT   8192
#define REL_THR 0.7f
#define REP_THR 0.7f

typedef __attribute__((ext_vector_type(2))) float v2f;
typedef __attribute__((ext_vector_type(8))) float v8f;

// ---------------- workspace layout (bytes) ----------------
#define OFF_REL   0u
#define OFF_REP   (OFF_REL  + (size_t)HW * 4u)
#define OFF_INVN  (OFF_REP  + (size_t)HW * 4u)
#define OFF_CS    (OFF_INVN + (size_t)HW * 4u)          // float[MAXCAND]
#define OFF_CI    (OFF_CS   + (size_t)MAXCAND * 4u)     // int[MAXCAND]
#define OFF_KEYS  (OFF_CI   + (size_t)MAXCAND * 4u)     // u64[NSORT]
#define OFF_CNT   (OFF_KEYS + (size_t)NSORT * 8u)       // int
#define OFF_HIST  (OFF_CNT  + 4u)                       // int[NBINS]
#define OFF_BASE  (OFF_HIST + (size_t)NBINS * 4u)       // int[NBINS]
#define OFF_FILL  (OFF_BASE + (size_t)NBINS * 4u)       // int[NBINS]
#define OFF_SEL   (OFF_FILL + (size_t)NBINS * 4u)       // int[NSORT]
#define ZERO_FROM OFF_KEYS
#define ZERO_LEN  (OFF_SEL - OFF_KEYS)

// =====================================================================
// Kernel 1: fused heads. One wave = one 16-pixel tile. GEMM via
// V_WMMA_F32_16X16X4_F32: A = f^2 (16 px x 4 ch), B = packed [4 x 16]
// weights with cols {clf0, clf1, sal0, sal1, ones, 0...}. The ones
// column yields sum(f^2) = ||f||^2 for the descriptor norm for free.
// =====================================================================
__global__ void heads_kernel(const float* __restrict__ f,
                             const float* __restrict__ clf_w,
                             const float* __restrict__ clf_b,
                             const float* __restrict__ sal_w,
                             const float* __restrict__ sal_b,
                             float* __restrict__ rel,
                             float* __restrict__ rep,
                             float* __restrict__ invn)
{
    __shared__ float Wlds[C_DIM * 16];      // packed B matrix [128][16]
    __shared__ float Dlds[8][16 * 5];       // per-wave 16px x 5 outputs

    for (int i = threadIdx.x; i < C_DIM * 16; i += blockDim.x) {
        int k = i >> 4, n = i & 15;
        float w = 0.0f;
        if (n < 2)      w = clf_w[k * 2 + n];
        else if (n < 4) w = sal_w[k * 2 + (n - 2)];
        else if (n == 4) w = 1.0f;
        Wlds[i] = w;
    }
    __syncthreads();

    const int wave = threadIdx.x >> 5;
    const int lane = threadIdx.x & 31;
    const int half = lane >> 4;             // ISA A layout: half 1 holds K+2,K+3
    const int col  = lane & 15;             // M (pixel) for A, N (output) for B/D
    const int pix0 = (blockIdx.x * 8 + wave) * 16;
    const int pix  = pix0 + col;

    v8f acc = {};
    #pragma unroll 4
    for (int k = 0; k < C_DIM; k += 4) {
        const int ka = k + half * 2;
        float a0 = f[(size_t)ka       * HW + pix];
        float a1 = f[(size_t)(ka + 1) * HW + pix];
        v2f A; A.x = a0 * a0; A.y = a1 * a1;
        v2f B; B.x = Wlds[ka * 16 + col];
               B.y = Wlds[(ka + 1) * 16 + col];
        acc = __builtin_amdgcn_wmma_f32_16x16x4_f32(
                  false, A, false, B, (short)0, acc, false, false);
    }

    // D tile: VGPR j = row M = j + half*8, col N = col (ISA 7.12.2)
    if (col < 5) {
        #pragma unroll
        for (int j = 0; j < 8; ++j)
            Dlds[wave][(j + half * 8) * 5 + col] = acc[j];
    }
    __syncthreads();

    if (lane < 16) {
        const float* d = &Dlds[wave][lane * 5];
        float u0 = d[0] + clf_b[0], u1 = d[1] + clf_b[1];
        float v0 = d[2] + sal_b[0], v1 = d[3] + sal_b[1];
        float ss = d[4];
        int p = pix0 + lane;
        rel[p]  = 1.0f / (1.0f + __expf(u0 - u1));   // softmax(.)[1]
        rep[p]  = 1.0f / (1.0f + __expf(v0 - v1));
        invn[p] = 1.0f / fmaxf(sqrtf(ss), 1e-12f);
    }
}

// =====================================================================
// Kernel 2: 3x3 NMS + dual threshold -> candidate compaction
// =====================================================================
__global__ void nms_kernel(const float* __restrict__ rel,
                           const float* __restrict__ rep,
                           float* __restrict__ cs, int* __restrict__ ci,
                           int* __restrict__ counter)
{
    int p = blockIdx.x * blockDim.x + threadIdx.x;
    if (p >= HW) return;
    float r  = rep[p];
    float rl = rel[p];
    if (r < REP_THR || rl < REL_THR) return;
    int y = p >> 10, x = p & (W_DIM - 1);
    float m = -INFINITY;
    #pragma unroll
    for (int dy = -1; dy <= 1; ++dy)
        #pragma unroll
        for (int dx = -1; dx <= 1; ++dx) {
            if (dy == 0 && dx == 0) continue;
            int ny = y + dy, nx = x + dx;
            if (ny < 0 || ny >= H_DIM || nx < 0 || nx >= W_DIM) continue;
            m = fmaxf(m, rep[ny * W_DIM + nx]);
        }
    if (r < m) return;                      // rep == maxpool3(rep)
    int slot = atomicAdd(counter, 1);
    if (slot < MAXCAND) { cs[slot] = rl * r; ci[slot] = p; }
}

// =====================================================================
// Kernel 3: score histogram (4096 bins over [0,1])
// =====================================================================
__global__ void hist_kernel(const float* __restrict__ cs,
                            const int* __restrict__ counter,
                            int* __restrict__ hist)
{
    int n = min(*counter, MAXCAND);
    int i = blockIdx.x * blockDim.x + threadIdx.x;
    if (i >= n) return;
    int bin = (int)(cs[i] * (float)NBINS);
    bin = max(0, min(NBINS - 1, bin));
    atomicAdd(&hist[bin], 1);
}

// =====================================================================
// Kernel 4: suffix scan of histogram -> base offset per bin
// =====================================================================
__global__ void scan_kernel(const int* __restrict__ hist, int* __restrict__ base)
{
    if (threadIdx.x == 0 && blockIdx.x == 0) {
        int running = 0;
        for (int b = NBINS - 1; b >= 0; --b) { base[b] = running; running += hist[b]; }
    }
}

// =====================================================================
// Kernel 5: bin-ordered compaction into 64-bit sort keys:
//   key = orderable(score) << 32 | (~pixel_idx)   (0 == empty slot)
// =====================================================================
__global__ void compact_kernel(const float* __restrict__ cs,
                               const int* __restrict__ ci,
                               const int* __restrict__ counter,
                               const int* __restrict__ base,
                               int* __restrict__ fill,
                               unsigned long long* __restrict__ keys)
{
    int n = min(*counter, MAXCAND);
    int i = blockIdx.x * blockDim.x + threadIdx.x;
    if (i >= n) return;
    float s = cs[i];
    int bin = max(0, min(NBINS - 1, (int)(s * (float)NBINS)));
    int pos = base[bin] + atomicAdd(&fill[bin], 1);
    if (pos >= NSORT) return;
    unsigned int u = __float_as_uint(s);
    u = (u & 0x80000000u) ? ~u : (u | 0x80000000u);      // order-preserving
    unsigned long long key = ((unsigned long long)u << 32) |
                             (unsigned long long)(0xFFFFFFFFu - (unsigned int)ci[i]);
    keys[pos] = key;
}

// =====================================================================
// Kernel 6: single-block bitonic sort (descending) of 8192 keys in
// 64 KB LDS (CDNA5 WGP has 320 KB), then emit keypoints/scores/sel_idx.
// =====================================================================
__global__ void __launch_bounds__(1024)
sort_write_kernel(const unsigned long long* __restrict__ keys,
                  float* __restrict__ keypoints,   // [K_OUT][2]
                  float* __restrict__ scores,      // [K_OUT]
                  int* __restrict__ sel_idx)       // [K_OUT]
{
    __shared__ unsigned long long sm[NSORT];        // 64 KB
    for (int i = threadIdx.x; i < NSORT; i += blockDim.x) sm[i] = keys[i];
    __syncthreads();

    for (unsigned int k = 2; k <= NSORT; k <<= 1) {
        for (unsigned int j = k >> 1; j > 0; j >>= 1) {
            for (unsigned int i = threadIdx.x; i < NSORT; i += blockDim.x) {
                unsigned int l = i ^ j;
                if (l > i) {
                    unsigned long long a = sm[i], b = sm[l];
                    bool desc = ((i & k) == 0);
                    if (desc ? (a < b) : (a > b)) { sm[i] = b; sm[l] = a; }
                }
            }
            __syncthreads();
        }
    }

    for (int r = threadIdx.x; r < K_OUT; r += blockDim.x) {
        unsigned long long key = sm[r];
        float score; int idx;
        if (key == 0ull) {                           // fewer than K candidates
            score = -INFINITY; idx = r;
        } else {
            unsigned int hi = (unsigned int)(key >> 32);
            hi = (hi & 0x80000000u) ? (hi ^ 0x80000000u) : ~hi;
            score = __uint_as_float(hi);
            idx = (int)(0xFFFFFFFFu - (unsigned int)(key & 0xFFFFFFFFu));
        }
        scores[r] = score;
        keypoints[2 * r + 0] = (float)(idx & (W_DIM - 1));  // x
        keypoints[2 * r + 1] = (float)(idx >> 10);          // y
        sel_idx[r] = idx;
    }
}

// =====================================================================
// Kernel 7: descriptor gather. One wave per keypoint, 4 channels/lane,
// scaled by the inv-norm map computed for free in the WMMA heads pass.
// =====================================================================
__global__ void gather_kernel(const float* __restrict__ f,
                              const float* __restrict__ invn,
                              const int* __restrict__ sel_idx,
                              float* __restrict__ desc)     // [K_OUT][C_DIM]
{
    int wave = threadIdx.x >> 5;
    int lane = threadIdx.x & 31;
    int kp = blockIdx.x * 8 + wave;
    if (kp >= K_OUT) return;
    int idx = sel_idx[kp];
    float in = invn[idx];
    #pragma unroll
    for (int j = 0; j < 4; ++j) {
        int c = lane + 32 * j;
        float v = f[(size_t)c * HW + idx];
        desc[(size_t)kp * C_DIM + c] = v * in;
    }
}

// =====================================================================
extern "C" void kernel_launch(void* const* d_in, const int* in_sizes, int n_in,
                              void* d_out, int out_size, void* d_ws, size_t ws_size,
                              hipStream_t stream)
{
    (void)in_sizes; (void)n_in; (void)out_size; (void)ws_size;

    const float* features = (const float*)d_in[0];
    const float* clf_w    = (const float*)d_in[1];
    const float* clf_b    = (const float*)d_in[2];
    const float* sal_w    = (const float*)d_in[3];
    const float* sal_b    = (const float*)d_in[4];
    // d_in[5] = max_keypoints (5000, matches K_OUT / out_size)

    char* ws = (char*)d_ws;
    float* rel   = (float*)(ws + OFF_REL);
    float* rep   = (float*)(ws + OFF_REP);
    float* invn  = (float*)(ws + OFF_INVN);
    float* cs    = (float*)(ws + OFF_CS);
    int*   ci    = (int*)  (ws + OFF_CI);
    unsigned long long* keys = (unsigned long long*)(ws + OFF_KEYS);
    int*   counter = (int*)(ws + OFF_CNT);
    int*   hist  = (int*)  (ws + OFF_HIST);
    int*   base  = (int*)  (ws + OFF_BASE);
    int*   fill  = (int*)  (ws + OFF_FILL);
    int*   sel   = (int*)  (ws + OFF_SEL);

    float* out_kp    = (float*)d_out;                 // [5000][2]
    float* out_score = out_kp + 2 * K_OUT;            // [5000]
    float* out_desc  = out_score + K_OUT;             // [5000][128]

    // zero keys + counter + hist + base + fill (fresh every call)
    hipMemsetAsync(ws + ZERO_FROM, 0, ZERO_LEN, stream);

    heads_kernel<<<HW / 128, 256, 0, stream>>>(features, clf_w, clf_b,
                                               sal_w, sal_b, rel, rep, invn);
    nms_kernel<<<HW / 256, 256, 0, stream>>>(rel, rep, cs, ci, counter);
    hist_kernel<<<MAXCAND / 256, 256, 0, stream>>>(cs, counter, hist);
    scan_kernel<<<1, 1, 0, stream>>>(hist, base);
    compact_kernel<<<MAXCAND / 256, 256, 0, stream>>>(cs, ci, counter, base, fill, keys);
    sort_write_kernel<<<1, 1024, 0, stream>>>(keys, out_kp, out_score, sel);
    gather_kernel<<<(K_OUT + 7) / 8, 256, 0, stream>>>(features, invn, sel, out_desc);
}